// MissingAwareTemporalAttention_14405320311486
// MI455X (gfx1250) — compile-verified
//
#include <hip/hip_runtime.h>
#include <hip/hip_bf16.h>

// Problem constants (match reference)
#define B_    256
#define T_    2048
#define HID_  128
#define ATT_  8

typedef __attribute__((ext_vector_type(2))) float v2f;
typedef __attribute__((ext_vector_type(8))) float v8f;
typedef __attribute__((ext_vector_type(4))) unsigned int u32x4;
typedef __attribute__((ext_vector_type(8))) int i32x8;
typedef __attribute__((ext_vector_type(4))) int i32x4;

constexpr int LDS_STRIDE     = 132;           // 128 + 4 pad -> conflict-free banks, 16B-aligned rows
constexpr int WAVES          = 4;             // wave32: 4 waves = 128 threads
constexpr int THREADS        = WAVES * 32;
constexpr int TILES          = T_ / 16;       // 128 tiles of 16 timesteps
constexpr int TILES_PER_WAVE = TILES / WAVES; // 32
constexpr int TILE_FLOATS    = 16 * LDS_STRIDE;

// ---------------------------------------------------------------------------
// TDM descriptor: load a 16x128 f32 tile (row stride 128 elements in memory)
// into LDS, inserting 4 DWORDs of padding after every 128 DWORDs (one row),
// i.e. LDS row stride = 132 floats -> bank-conflict-free WMMA fragment reads.
//   group0: count=1, lds_addr, global_addr(57b), type=2
//   group1: data_size=4B, pad_enable, pad_interval=6 (128 dw), pad_amount=3
//           (4 dw), tensor_dim0=128, tensor_dim1=rows, tile=128x16, stride=128
// ---------------------------------------------------------------------------
__device__ __forceinline__ void tdm_load_tile(const float* gsrc, unsigned lds_byte,
                                              unsigned rows_avail) {
  const unsigned long long ga = (unsigned long long)(uintptr_t)gsrc;
  u32x4 g0;
  g0[0] = 1u;                                            // count=1, user mode
  g0[1] = lds_byte;                                      // LDS byte address
  g0[2] = (unsigned)ga;                                  // global_addr[31:0]
  g0[3] = (unsigned)((ga >> 32) & 0x01FFFFFFu) | (2u << 30); // [56:32] | type=2
  i32x8 g1;
  g1[0] = (int)((2u << 16)        // data_size = 4 bytes
              | (1u << 20)        // pad_enable
              | (6u << 22)        // pad_interval: 128 DWORDs
              | (3u << 25));      // pad_amount: 4 DWORDs
  g1[1] = (int)(128u << 16);                  // tensor_dim0 = 128 (lo16 @ [31:16])
  g1[2] = (int)((rows_avail & 0xFFFFu) << 16);// tensor_dim1 lo16 @ [31:16]
  g1[3] = (int)((rows_avail >> 16) & 0xFFFFu) | (int)(128u << 16); // dim1 hi | tile_dim0=128
  g1[4] = 16;                                 // tile_dim1 = 16 rows
  g1[5] = 128;                                // tensor_dim0_stride = 128 elems
  g1[6] = 0;
  g1[7] = 0;
  const i32x4 gz4 = {0, 0, 0, 0};             // groups 2/3 unused (2D tensor)
  const i32x8 gz8 = {0, 0, 0, 0, 0, 0, 0, 0};
  __builtin_amdgcn_tensor_load_to_lds(g0, g1, gz4, gz4, gz8, 0);
}

// ---------------------------------------------------------------------------
// Kernel 1: per-batch projected query vector  w[b,h] = sum_a Wx[h,a] * q[b,a],
//           q[b,a] = sum_h H[b,T-1,h] * Wt[h,a].   (dot[b,t] == H[b,t]·w[b])
// ---------------------------------------------------------------------------
__global__ __launch_bounds__(HID_)
void matar_w_kernel(const float* __restrict__ H,
                    const float* __restrict__ Wt,
                    const float* __restrict__ Wx,
                    float* __restrict__ w) {
  __shared__ float hl[HID_];
  __shared__ float q[ATT_];
  const int b   = blockIdx.x;
  const int tid = threadIdx.x;
  hl[tid] = H[((size_t)b * T_ + (T_ - 1)) * HID_ + tid];
  __syncthreads();
  if (tid < ATT_) {
    float acc = 0.f;
    for (int h = 0; h < HID_; ++h) acc += hl[h] * Wt[h * ATT_ + tid];
    q[tid] = acc;
  }
  __syncthreads();
  float wv = 0.f;
#pragma unroll
  for (int a = 0; a < ATT_; ++a) wv += Wx[tid * ATT_ + a] * q[a];
  w[b * HID_ + tid] = wv;
}

// ---------------------------------------------------------------------------
// Kernel 2: fused single-pass-over-H attention (flash-style online softmax).
// One block per batch; each wave owns double-buffered TDM-loaded 16x128 tiles
// reused for (a) the WMMA dot and (b) the online-softmax v accumulation.
// ---------------------------------------------------------------------------
__global__ __launch_bounds__(THREADS)
void matar_main_kernel(const float* __restrict__ H,
                       const float* __restrict__ mask,
                       const float* __restrict__ wvec,
                       const float* __restrict__ rate,
                       const float* __restrict__ obs_bias_p,
                       const float* __restrict__ miss_bias_p,
                       float* __restrict__ out_v,
                       float* __restrict__ out_a) {
  __shared__ float ldsA[2 * WAVES * TILE_FLOATS]; // double-buffered staging, 67.5 KB
  __shared__ float eLds[T_];                      // 8 KB: e+bias per timestep
  __shared__ float wLds[HID_];
  __shared__ float vsum[HID_];
  __shared__ float mW[WAVES], sW[WAVES];

  const int b    = blockIdx.x;
  const int tid  = threadIdx.x;
  const int wid  = tid >> 5;
  const int lane = tid & 31;
  const int mrow = lane & 15;   // M index for WMMA A fragment
  const int hi   = lane >> 4;   // upper half-wave holds K+2 / M+8 per ISA layout

  if (tid < HID_) { wLds[tid] = wvec[b * HID_ + tid]; vsum[tid] = 0.f; }
  const float sig_rate = 1.f / (1.f + __expf(-rate[0]));
  const float obs_b    = obs_bias_p[0];
  const float miss_b   = miss_bias_p[0];
  __syncthreads();

  float* buf0 = &ldsA[(0 * WAVES + wid) * TILE_FLOATS];
  float* buf1 = &ldsA[(1 * WAVES + wid) * TILE_FLOATS];
  const unsigned lb0 = (unsigned)(uintptr_t)(void*)buf0; // LDS byte offset
  const unsigned lb1 = (unsigned)(uintptr_t)(void*)buf1;
  const float* Hb = H + (size_t)b * T_ * HID_;

  float m_run = -3.4e38f;   // running max (online softmax)
  float s_run = 0.f;        // running sum
  float vacc0 = 0.f, vacc1 = 0.f, vacc2 = 0.f, vacc3 = 0.f; // h = 4*lane..+3

  // prologue: start DMA of tile 0 into buffer 0
  {
    const int t0 = wid * 16;
    tdm_load_tile(Hb + (size_t)t0 * HID_, lb0, (unsigned)(T_ - t0));
  }

  for (int ti = 0; ti < TILES_PER_WAVE; ++ti) {
    const int   t0   = (wid + WAVES * ti) * 16;
    float*      tile = (ti & 1) ? buf1 : buf0;

    // Pending DS reads of the *other* buffer must retire before the TDM engine
    // overwrites it (counters do not order TDM writes vs DS reads).
    asm volatile("s_wait_dscnt 0" ::: "memory");
    if (ti + 1 < TILES_PER_WAVE) {
      const int t0n = (wid + WAVES * (ti + 1)) * 16;
      tdm_load_tile(Hb + (size_t)t0n * HID_, ((ti + 1) & 1) ? lb1 : lb0,
                    (unsigned)(T_ - t0n));
      __builtin_amdgcn_s_wait_tensorcnt(1);  // current tile complete
    } else {
      __builtin_amdgcn_s_wait_tensorcnt(0);
    }

    // ---- WMMA: D[m][n] = sum_k Htile[m][k] * w[k]  (w replicated over N) ----
    v8f acc = {0.f, 0.f, 0.f, 0.f, 0.f, 0.f, 0.f, 0.f};
#pragma unroll
    for (int k0 = 0; k0 < HID_; k0 += 4) {
      const int ka = k0 + 2 * hi;                 // A layout: lanes 16-31 hold K+2
      v2f af, bf;
      af.x = tile[mrow * LDS_STRIDE + ka];
      af.y = tile[mrow * LDS_STRIDE + ka + 1];
      bf.x = wLds[ka];                            // B[k][n] = w[k] for all n
      bf.y = wLds[ka + 1];
      acc = __builtin_amdgcn_wmma_f32_16x16x4_f32(
          false, af, false, bf, (short)0, acc, false, false);
    }

    // ---- e = relu(s/denom) + bias for this tile's 16 rows ----
    // C/D layout: lane<16 -> rows t0+0..7 in acc[0..7]; lane>=16 -> rows t0+8..15
    float own[8];
#pragma unroll
    for (int r = 0; r < 8; ++r) {
      const int   trow  = t0 + r + 8 * hi;
      const float dot   = acc[r];
      const float s     = 1.f / (1.f + __expf(-dot));
      const float btime = (float)(T_ - trow);     // reversed arange + 1
      const float denom = sig_rate * (__logf(2.72f + (1.f - s)) * btime);
      float e = fmaxf(s / denom, 0.f);
      e += (mask[(size_t)b * T_ + trow] > 0.5f) ? obs_b : miss_b;
      own[r] = e;
      if (mrow == 0) eLds[trow] = e;              // lanes 0 and 16 record e
    }

    // exchange halves so every lane holds all 16 e values
    float pv[16];
#pragma unroll
    for (int r = 0; r < 8; ++r) {
      const float oth = __shfl_xor(own[r], 16, 32);
      pv[r + 8 * hi]       = own[r];
      pv[r + 8 * (1 - hi)] = oth;
    }
    float tmax = pv[0];
#pragma unroll
    for (int r = 1; r < 16; ++r) tmax = fmaxf(tmax, pv[r]);

    // ---- online softmax rescale + v accumulation (reads tile from LDS) ----
    const float m_new = fmaxf(m_run, tmax);
    const float c     = __expf(m_run - m_new);
    s_run *= c; vacc0 *= c; vacc1 *= c; vacc2 *= c; vacc3 *= c;
#pragma unroll
    for (int r = 0; r < 16; ++r) {
      const float  p  = __expf(pv[r] - m_new);
      s_run += p;
      const float4 hv = *(const float4*)(tile + r * LDS_STRIDE + 4 * lane);
      vacc0 += p * hv.x; vacc1 += p * hv.y; vacc2 += p * hv.z; vacc3 += p * hv.w;
    }
    m_run = m_new;
  }

  // ---- combine the 4 waves' (m, s, vacc) ----
  if (lane == 0) { mW[wid] = m_run; sW[wid] = s_run; }
  __syncthreads();
  float M = mW[0];
#pragma unroll
  for (int wv = 1; wv < WAVES; ++wv) M = fmaxf(M, mW[wv]);
  float S = 0.f;
#pragma unroll
  for (int wv = 0; wv < WAVES; ++wv) S += sW[wv] * __expf(mW[wv] - M);

  const float myscale = __expf(m_run - M);
  for (int wv = 0; wv < WAVES; ++wv) {        // uniform loop; guarded writes
    if (wid == wv) {
      vsum[4 * lane + 0] += vacc0 * myscale;
      vsum[4 * lane + 1] += vacc1 * myscale;
      vsum[4 * lane + 2] += vacc2 * myscale;
      vsum[4 * lane + 3] += vacc3 * myscale;
    }
    __syncthreads();
  }

  // ---- outputs: a = exp(e - M)/S  (coalesced),  v = vsum/S ----
#pragma unroll
  for (int i = 0; i < T_ / THREADS; ++i) {
    const int t = tid + THREADS * i;
    out_a[(size_t)b * T_ + t] = __expf(eLds[t] - M) / S;
  }
  if (tid < HID_) out_v[b * HID_ + tid] = vsum[tid] / S;
}

// ---------------------------------------------------------------------------
extern "C" void kernel_launch(void* const* d_in, const int* in_sizes, int n_in,
                              void* d_out, int out_size, void* d_ws, size_t ws_size,
                              hipStream_t stream) {
  const float* H    = (const float*)d_in[0];
  const float* mask = (const float*)d_in[1];
  const float* Wt   = (const float*)d_in[2];
  const float* Wx   = (const float*)d_in[3];
  const float* rate = (const float*)d_in[4];
  const float* obsb = (const float*)d_in[5];
  const float* misb = (const float*)d_in[6];

  float* out   = (float*)d_out;
  float* out_v = out;                 // [B, HID] first
  float* out_a = out + B_ * HID_;     // then [B, T]
  float* w     = (float*)d_ws;        // [B, HID] scratch

  matar_w_kernel<<<B_, HID_, 0, stream>>>(H, Wt, Wx, w);
  matar_main_kernel<<<B_, THREADS, 0, stream>>>(H, mask, w, rate, obsb, misb,
                                                out_v, out_a);
}